// CrossOpLayerUTPM_17875653886245
// MI455X (gfx1250) — compile-verified
//
#include <hip/hip_runtime.h>
#include <hip/hip_bf16.h>
#include <stdint.h>

// Problem constants (reference: B=4096, N=256, K=16)
#define NB    4096
#define NN    256
#define NK    16
#define PTOT  32640          // N*(N-1)/2

typedef __attribute__((ext_vector_type(2))) float    v2f;
typedef __attribute__((ext_vector_type(4))) float    v4f;
typedef __attribute__((ext_vector_type(8))) float    v8f;
typedef __attribute__((ext_vector_type(4))) uint32_t v4u;

// ---------------------------------------------------------------------------
// Kernel 1: S = L * L^T  (256x16 @ 16x256 -> 256x256) via V_WMMA_F32_16X16X4_F32
// One wave32 computes one 16x16 tile of S; K=16 -> 4 chained WMMAs.
// f32 WMMA keeps full fp32 precision (matches the fp32 reference).
// ---------------------------------------------------------------------------
__global__ __launch_bounds__(32)
void gram_wmma_f32(const float* __restrict__ L, float* __restrict__ S) {
    const int tile = blockIdx.x;          // 0..255 -> (I,J) 16x16 tile grid
    const int I = tile >> 4;
    const int J = tile & 15;
    const int lane = threadIdx.x;         // wave32
    const int half = lane >> 4;           // 0: K pair {k,k+1}; 1: {k+2,k+3}
    const int l15  = lane & 15;

    v8f c = {};
    #pragma unroll
    for (int k0 = 0; k0 < NK; k0 += 4) {
        // A (16x4, M x K): lane l<16 holds (K=k0,k0+1) of row M=l; lanes 16-31 K=k0+2,k0+3
        const float* arow = L + (I * 16 + l15) * NK + k0 + half * 2;
        v2f a; a.x = arow[0]; a.y = arow[1];
        // B (4x16, K x N) = L^T tile: B[k][n] = L[J*16+n][k]; mirrored lane layout
        const float* brow = L + (J * 16 + l15) * NK + k0 + half * 2;
        v2f b; b.x = brow[0]; b.y = brow[1];
        c = __builtin_amdgcn_wmma_f32_16x16x4_f32(
                /*neg_a=*/false, a, /*neg_b=*/false, b,
                /*c_mod=*/(short)0, c, /*reuse_a=*/false, /*reuse_b=*/false);
    }
    // C/D layout: VGPR r, lanes 0-15 -> (M=r, N=lane); lanes 16-31 -> (M=r+8, N=lane-16)
    const int n = l15;
    const int mbase = half * 8;
    #pragma unroll
    for (int r = 0; r < 8; ++r) {
        S[(I * 16 + mbase + r) * NN + J * 16 + n] = c[r];
    }
}

// ---------------------------------------------------------------------------
// Kernel 2: build spair[p] = S[i,j] and packed idx[p] = i | (j<<16) for the
// strict upper triangle in jnp.triu_indices (row-major) order.
// ---------------------------------------------------------------------------
__global__ __launch_bounds__(256)
void build_pairs(const float* __restrict__ S,
                 float* __restrict__ spair,
                 uint32_t* __restrict__ idx) {
    const int i = blockIdx.x;             // 0..254
    const int j = threadIdx.x;            // 0..255
    if (j <= i) return;
    const int start = i * (2 * NN - i - 1) / 2;   // i*(511-i)/2
    const int p = start + (j - i - 1);
    spair[p] = S[i * NN + j];
    idx[p]   = (uint32_t)i | ((uint32_t)j << 16);
}

// ---------------------------------------------------------------------------
// Kernel 3 (the bandwidth kernel): out[b,p] = x[b,i]*x[b,j]*spair[p].
// Block = 256 threads; owns 8 b-rows (x rows cached in 8KB LDS) and a 1024-pair
// chunk. Each thread loads its pair metadata once (v4f + v4u from L2,
// reused across 8 b's) and emits coalesced non-temporal 128-bit stores.
// ---------------------------------------------------------------------------
__global__ __launch_bounds__(256)
void cross_utpm_main(const float* __restrict__ x,
                     const float* __restrict__ spair,
                     const uint32_t* __restrict__ idx,
                     float* __restrict__ out) {
    __shared__ float xs[8][NN];

    const int bBase = blockIdx.y * 8;
    const int pBase = blockIdx.x * 1024 + threadIdx.x * 4;

    // Cooperative load: 8 rows * 256 floats = 2048 floats by 256 threads.
    #pragma unroll
    for (int t = threadIdx.x; t < 8 * NN; t += 256) {
        xs[t >> 8][t & (NN - 1)] = x[(size_t)(bBase + (t >> 8)) * NN + (t & (NN - 1))];
    }
    __syncthreads();

    if (pBase >= PTOT) return;            // last p-chunk is partial (896 of 1024)

    const v4f sp = *(const v4f*)(spair + pBase);
    const v4u id = *(const v4u*)(idx + pBase);
    const int i0 = id.x & 0xFFFF, j0 = id.x >> 16;
    const int i1 = id.y & 0xFFFF, j1 = id.y >> 16;
    const int i2 = id.z & 0xFFFF, j2 = id.z >> 16;
    const int i3 = id.w & 0xFFFF, j3 = id.w >> 16;

    #pragma unroll
    for (int bb = 0; bb < 8; ++bb) {
        const float* __restrict__ xr = xs[bb];
        v4f r;
        r.x = xr[i0] * xr[j0] * sp.x;
        r.y = xr[i1] * xr[j1] * sp.y;
        r.z = xr[i2] * xr[j2] * sp.z;
        r.w = xr[i3] * xr[j3] * sp.w;
        v4f* dst = (v4f*)(out + (size_t)(bBase + bb) * PTOT + pBase);
        __builtin_nontemporal_store(r, dst);   // write-once stream: keep L2 for x/spair/idx
    }
}

// ---------------------------------------------------------------------------
// Launcher
// ---------------------------------------------------------------------------
extern "C" void kernel_launch(void* const* d_in, const int* in_sizes, int n_in,
                              void* d_out, int out_size, void* d_ws, size_t ws_size,
                              hipStream_t stream) {
    const float* x = (const float*)d_in[0];        // [4096, 256]
    const float* L = (const float*)d_in[1];        // [256, 16]
    float* out = (float*)d_out;                    // [4096, 32640]

    // Workspace layout (all 16B aligned): S | spair | idx
    char* ws = (char*)d_ws;
    float*    S     = (float*)ws;                            // 256*256*4 = 262144 B
    float*    spair = (float*)(ws + 262144);                 // 32640*4  = 130560 B
    uint32_t* idx   = (uint32_t*)(ws + 262144 + 130560);     // 32640*4  = 130560 B

    // 1) tiny gram via f32 WMMA: 256 tiles, one wave each
    gram_wmma_f32<<<256, 32, 0, stream>>>(L, S);

    // 2) pair gather/index tables (L2-resident metadata)
    build_pairs<<<255, 256, 0, stream>>>(S, spair, idx);

    // 3) streaming kernel: 32 p-chunks x 512 b-chunks = 16384 blocks
    dim3 grid(32, NB / 8);
    cross_utpm_main<<<grid, 256, 0, stream>>>(x, spair, idx, out);
}